// PairwiseRankingLoss_48687749268113
// MI455X (gfx1250) — compile-verified
//
#include <hip/hip_runtime.h>

#define N_TOTAL 16384
#define BLOCK   256
#define JTILE   2048
#define NWAVES  (BLOCK / 32)

typedef __attribute__((ext_vector_type(2))) float v2f;
typedef __attribute__((ext_vector_type(8))) float v8f;

// d_ws layout: [0..7] double total, [8..15] u64 pair count
__global__ void prl_init_kernel(double* total, unsigned long long* cnt) {
    *total = 0.0;
    *cnt   = 0ull;
}

__global__ __launch_bounds__(BLOCK)
void prl_pair_kernel(const float* __restrict__ scores,
                     const int*   __restrict__ ranks,
                     double*      __restrict__ total,
                     unsigned long long* __restrict__ count) {
    // (score, rank) packed per j: one ds_load_b64 per pair in the hot loop.
    __shared__ int2     srj[JTILE];
    __shared__ float    wsum[NWAVES];
    __shared__ unsigned wcnt[NWAVES];

    const int tid = threadIdx.x;
    const int j0  = blockIdx.y * JTILE;

    // Stage this block's j-tile in LDS (coalesced global loads, 8 per thread).
    for (int k = tid; k < JTILE; k += BLOCK) {
        srj[k] = make_int2(__float_as_int(scores[j0 + k]), ranks[j0 + k]);
    }
    __syncthreads();

    const int   i  = blockIdx.x * BLOCK + tid;   // one i-row per thread
    const float si = scores[i];
    const int   ri = ranks[i];

    // Per-lane partials over <=2048 terms: exact-enough in f32 (count is exact).
    float acc  = 0.0f;
    float cntf = 0.0f;
#pragma unroll 8
    for (int k = 0; k < JTILE; ++k) {
        const int2  v = srj[k];                       // single b64 broadcast read
        const float d = 1.0f - si + __int_as_float(v.x);
        const float h = fmaxf(d, 0.0f);
        const bool  m = ri < v.y;                     // rank_i < rank_j
        acc  += m ? h : 0.0f;
        cntf += m ? 1.0f : 0.0f;
    }

    // ---- Dual wave32 reduction (loss sum + pair count) in ONE fp32 WMMA ----
    // A (16x4): K=0/2 hold acc, K=1/3 hold cntf (lanes 0-15 -> K=0,1; 16-31 -> K=2,3).
    // B (4x16) column selector: cols n<8 pick acc rows (K=0,2), cols n>=8 pick cnt rows.
    //   => D[m][n<8]  = acc(m)  + acc(m+16)
    //      D[m][n>=8] = cntf(m) + cntf(m+16)
    const int  lane = tid & 31;
    const bool losel = (lane & 15) < 8;
    v2f a; a[0] = acc;               a[1] = cntf;
    v2f b; b[0] = losel ? 1.f : 0.f; b[1] = losel ? 0.f : 1.f;
    v8f c = {};
    c = __builtin_amdgcn_wmma_f32_16x16x4_f32(
            /*neg_a=*/false, a, /*neg_b=*/false, b,
            /*c_mod=*/(short)0, c, /*reuse_a=*/false, /*reuse_b=*/false);
    // Lane n<16 holds D[0..7][n]; lane n>=16 holds D[8..15][n-16].
    float t = c[0] + c[1] + c[2] + c[3] + c[4] + c[5] + c[6] + c[7];
    t += __shfl_xor(t, 16, 32);   // lane 0: full acc total; lane 8: full cnt total

    const int wave = tid >> 5;
    if (lane == 0) wsum[wave] = t;
    if (lane == 8) wcnt[wave] = (unsigned)t;   // exact: <= 32*2048 < 2^24
    __syncthreads();

    if (tid == 0) {
        float bs = 0.0f;
        unsigned long long bc = 0ull;
#pragma unroll
        for (int w = 0; w < NWAVES; ++w) { bs += wsum[w]; bc += wcnt[w]; }
        atomicAdd(total, (double)bs);          // global_atomic_add_f64
        atomicAdd(count, bc);                  // global_atomic_add_u64
    }
}

__global__ void prl_final_kernel(const double* total,
                                 const unsigned long long* cnt,
                                 float* out) {
    const unsigned long long c = *cnt;
    out[0] = (c > 0ull) ? (float)(*total / (double)c) : 0.0f;
}

extern "C" void kernel_launch(void* const* d_in, const int* in_sizes, int n_in,
                              void* d_out, int out_size, void* d_ws, size_t ws_size,
                              hipStream_t stream) {
    const float* scores = (const float*)d_in[0];
    const int*   ranks  = (const int*)d_in[1];

    double*             total = (double*)d_ws;
    unsigned long long* count = (unsigned long long*)((char*)d_ws + 8);

    prl_init_kernel<<<1, 1, 0, stream>>>(total, count);

    dim3 grid(N_TOTAL / BLOCK, N_TOTAL / JTILE);   // (64, 8) = 512 workgroups
    prl_pair_kernel<<<grid, BLOCK, 0, stream>>>(scores, ranks, total, count);

    prl_final_kernel<<<1, 1, 0, stream>>>(total, count, (float*)d_out);
}